// HeteroGCN_84988812853627
// MI455X (gfx1250) — compile-verified
//
#include <hip/hip_runtime.h>

typedef float v2f __attribute__((ext_vector_type(2)));
typedef float v8f __attribute__((ext_vector_type(8)));

#define TPB 256

// ---------------------------------------------------------------- init
// Zero degree counters and layer-1 accumulator; seed out with sum of b2.
__global__ __launch_bounds__(TPB) void hg_init(unsigned* __restrict__ deg,
                                               float* __restrict__ acc1,
                                               float* __restrict__ out,
                                               const float* __restrict__ b2,
                                               int N) {
    long i = (long)blockIdx.x * TPB + threadIdx.x;   // spans 16*N
    if (i < 6L * N)  deg[i] = 0u;
    if (i < 16L * N) acc1[i] = 0.0f;
    if (i < N)       out[i] = b2[0] + b2[1] + b2[2];
}

// ---------------------------------------------------------------- degrees
__global__ __launch_bounds__(TPB) void hg_degree(const int* __restrict__ src,
                                                 const int* __restrict__ dst,
                                                 int E,
                                                 unsigned* __restrict__ dout,
                                                 unsigned* __restrict__ din) {
    int e = blockIdx.x * TPB + threadIdx.x;
    if (e < E) {
        atomicAdd(&dout[src[e]], 1u);
        atomicAdd(&din[dst[e]], 1u);
    }
}

// Convert u32 counts -> rsqrt(max(deg,1)) in place.
__global__ __launch_bounds__(TPB) void hg_scale(unsigned* __restrict__ deg, int n) {
    int i = blockIdx.x * TPB + threadIdx.x;
    if (i < n) {
        float d = (float)deg[i];
        if (d < 1.0f) d = 1.0f;
        ((float*)deg)[i] = rsqrtf(d);
    }
}

// ---------------------------------------------------------------- layer-1 GEMM
// h_pre[i,:] = (x[i,:] * s_out[i]) @ W   with W [32 x 16] row-major.
// One wave32 per 16-row tile; K=32 via 8 chained V_WMMA_F32_16X16X4_F32.
__global__ __launch_bounds__(TPB) void hg_gemm1_wmma(const float* __restrict__ x,
                                                     const float* __restrict__ s_out,
                                                     const float* __restrict__ W,
                                                     float* __restrict__ h_pre,
                                                     int N) {
    const int lane = threadIdx.x & 31;
    const int wave = threadIdx.x >> 5;
    const int tile = blockIdx.x * (TPB / 32) + wave;
    const int row0 = tile * 16;
    if (row0 >= N) return;                 // wave-uniform: EXEC stays all-1s otherwise

    const int half = lane >> 4;            // 0: K 0/1 of each 4-block, 1: K 2/3
    const int lm   = lane & 15;            // row (A) / column (B,D) within tile
    int row  = row0 + lm;
    int rowc = row < N ? row : N - 1;      // clamp loads at the ragged edge
    const float sc = s_out[rowc];

    v8f c = {};
#pragma unroll
    for (int kb = 0; kb < 8; ++kb) {
        const int k0 = kb * 4 + half * 2;
        v2f a, b;
        a.x = x[(size_t)rowc * 32 + k0]     * sc;
        a.y = x[(size_t)rowc * 32 + k0 + 1] * sc;
        b.x = W[(k0)     * 16 + lm];
        b.y = W[(k0 + 1) * 16 + lm];
        c = __builtin_amdgcn_wmma_f32_16x16x4_f32(
                /*neg_a=*/false, a, /*neg_b=*/false, b,
                /*c_mod=*/(short)0, c, /*reuse_a=*/false, /*reuse_b=*/false);
    }

    // D: VGPR v holds rows (row0+v, lanes 0-15) and (row0+v+8, lanes 16-31),
    // column = lm. Fast path when the whole tile is in range (wave-uniform
    // branch -> no per-store EXEC churn); N % 16 == 0 here so it always hits.
    float* __restrict__ dp = h_pre + (size_t)(row0 + half * 8) * 16 + lm;
    if (row0 + 16 <= N) {
#pragma unroll
        for (int v = 0; v < 8; ++v) dp[(size_t)v * 16] = c[v];
    } else {
#pragma unroll
        for (int v = 0; v < 8; ++v) {
            const int orow = row0 + v + half * 8;
            if (orow < N) h_pre[(size_t)orow * 16 + lm] = c[v];
        }
    }
}

// ---------------------------------------------------------------- layer-1 scatter
// acc1[dst,:] += h_pre[src,:] * s_in[dst]   (scale folded in => one accumulator
// shared by all relations). One thread per edge, 4x float4 loads, 16 atomics.
__global__ __launch_bounds__(TPB) void hg_scatter1(const int* __restrict__ src,
                                                   const int* __restrict__ dst,
                                                   int E,
                                                   const float* __restrict__ h_pre,
                                                   const float* __restrict__ s_in,
                                                   float* __restrict__ acc) {
    int e = blockIdx.x * TPB + threadIdx.x;
    if (e >= E) return;
    const int s = src[e], d = dst[e];
    const float sc = s_in[d];
    const float4* hp = (const float4*)(h_pre + (size_t)s * 16);
    float* ap = acc + (size_t)d * 16;
#pragma unroll
    for (int q = 0; q < 4; ++q) {
        float4 v = hp[q];
        atomicAdd(ap + q * 4 + 0, v.x * sc);
        atomicAdd(ap + q * 4 + 1, v.y * sc);
        atomicAdd(ap + q * 4 + 2, v.z * sc);
        atomicAdd(ap + q * 4 + 3, v.w * sc);
    }
}

// acc = relu(acc + b1[0]+b1[1]+b1[2]) in place -> becomes h.
__global__ __launch_bounds__(TPB) void hg_relu_bias(float* __restrict__ acc,
                                                    const float* __restrict__ b1,
                                                    long n16) {
    long i = (long)blockIdx.x * TPB + threadIdx.x;
    if (i < n16) {
        int f = (int)(i & 15);
        float v = acc[i] + b1[f] + b1[16 + f] + b1[32 + f];
        acc[i] = v > 0.0f ? v : 0.0f;
    }
}

// ---------------------------------------------------------------- layer-2
// z[i] = s_out[i] * dot(h[i,:], w[0:16])   (Fout==1: plain VALU dot product)
__global__ __launch_bounds__(TPB) void hg_gemm2(const float* __restrict__ h,
                                                const float* __restrict__ s_out,
                                                const float* __restrict__ w,
                                                float* __restrict__ z,
                                                int N) {
    int i = blockIdx.x * TPB + threadIdx.x;
    if (i >= N) return;
    const float4* hp = (const float4*)(h + (size_t)i * 16);
    const float4* wp = (const float4*)w;   // 64B-aligned (r*16 floats)
    float acc = 0.0f;
#pragma unroll
    for (int q = 0; q < 4; ++q) {
        float4 a = hp[q], b = wp[q];
        acc += a.x * b.x + a.y * b.y + a.z * b.z + a.w * b.w;
    }
    z[i] = acc * s_out[i];
}

__global__ __launch_bounds__(TPB) void hg_scatter2(const int* __restrict__ src,
                                                   const int* __restrict__ dst,
                                                   int E,
                                                   const float* __restrict__ z,
                                                   const float* __restrict__ s_in,
                                                   float* __restrict__ out) {
    int e = blockIdx.x * TPB + threadIdx.x;
    if (e < E) {
        int d = dst[e];
        atomicAdd(&out[d], z[src[e]] * s_in[d]);
    }
}

// ================================================================ driver
extern "C" void kernel_launch(void* const* d_in, const int* in_sizes, int n_in,
                              void* d_out, int out_size, void* d_ws, size_t ws_size,
                              hipStream_t stream) {
    const float* x  = (const float*)d_in[0];
    const int* src[3] = {(const int*)d_in[1], (const int*)d_in[3], (const int*)d_in[5]};
    const int* dst[3] = {(const int*)d_in[2], (const int*)d_in[4], (const int*)d_in[6]};
    const int  E[3]   = {in_sizes[1], in_sizes[3], in_sizes[5]};
    const float* W1 = (const float*)d_in[7];   // [3,32,16]
    const float* b1 = (const float*)d_in[8];   // [3,16]
    const float* W2 = (const float*)d_in[9];   // [3,16,1]
    const float* b2 = (const float*)d_in[10];  // [3,1]
    const int N = in_sizes[0] / 32;
    float* out = (float*)d_out;

    // ---- workspace carve-up (≈7.8 MB) ----
    char* ws = (char*)d_ws;
    unsigned* deg = (unsigned*)ws;                         // 6*N u32 -> becomes scales
    float* scales = (float*)ws;                            // s_out_r = scales+r*N ; s_in_r = scales+(3+r)*N
    float* acc1   = (float*)(ws + (size_t)6 * N * 4);      // [N,16]
    float* hpre   = acc1 + (size_t)N * 16;                 // [N,16] (reused per relation)
    float* z      = hpre + (size_t)N * 16;                 // [N]    (reused per relation)

    // ---- init ----
    {
        long span = 16L * N;
        hg_init<<<dim3((unsigned)((span + TPB - 1) / TPB)), dim3(TPB), 0, stream>>>(
            deg, acc1, out, b2, N);
    }

    // ---- degrees per relation ----
    for (int r = 0; r < 3; ++r) {
        hg_degree<<<dim3((E[r] + TPB - 1) / TPB), dim3(TPB), 0, stream>>>(
            src[r], dst[r], E[r], deg + (size_t)r * N, deg + (size_t)(3 + r) * N);
    }
    hg_scale<<<dim3((6 * N + TPB - 1) / TPB), dim3(TPB), 0, stream>>>(deg, 6 * N);

    // ---- layer 1: per relation, WMMA transform then fused-scale scatter ----
    const int tiles  = (N + 15) / 16;
    const int gblk   = (tiles + (TPB / 32) - 1) / (TPB / 32);
    for (int r = 0; r < 3; ++r) {
        hg_gemm1_wmma<<<dim3(gblk), dim3(TPB), 0, stream>>>(
            x, scales + (size_t)r * N, W1 + (size_t)r * 32 * 16, hpre, N);
        hg_scatter1<<<dim3((E[r] + TPB - 1) / TPB), dim3(TPB), 0, stream>>>(
            src[r], dst[r], E[r], hpre, scales + (size_t)(3 + r) * N, acc1);
    }
    {
        long n16 = 16L * N;
        hg_relu_bias<<<dim3((unsigned)((n16 + TPB - 1) / TPB)), dim3(TPB), 0, stream>>>(
            acc1, b1, n16);
    }

    // ---- layer 2: per relation, dot-product transform then scalar scatter ----
    for (int r = 0; r < 3; ++r) {
        hg_gemm2<<<dim3((N + TPB - 1) / TPB), dim3(TPB), 0, stream>>>(
            acc1, scales + (size_t)r * N, W2 + (size_t)r * 16, z, N);
        hg_scatter2<<<dim3((E[r] + TPB - 1) / TPB), dim3(TPB), 0, stream>>>(
            src[r], dst[r], E[r], z, scales + (size_t)(3 + r) * N, out);
    }
}